// SelfAttention_15298673509053
// MI455X (gfx1250) — compile-verified
//
#include <hip/hip_runtime.h>
#include <math.h>

// Problem constants (reference: B=128, S=1024, PATCH=KEY=16)
#define BB 128
#define SS 1024
#define DD 16

// All exponentials are done in the exp2 domain (v_exp_f32 is natively exp2):
// logits are scaled once by (1/3)*log2(e) straight out of the WMMA accumulator.
#define SCALE 0.48089834696298783f // log2(e) / 3

// Hardware exp2: lowers directly to v_exp_f32 (TRANS pipe), no wrapper.
#define EXP2F(x) __builtin_amdgcn_exp2f(x)

typedef __attribute__((ext_vector_type(2))) float v2f;
typedef __attribute__((ext_vector_type(8))) float v8f;

#define LDSP 17 // padded LDS row stride (floats): 17 coprime with 64 banks -> conflict-free

// Stage one 16x16 f32 tile (row-major, contiguous 16-float rows) from global into LDS.
// 32 lanes: lane loads 8 consecutive floats (2x float4, fully coalesced 64B rows).
__device__ __forceinline__ void stage_tile(const float* __restrict__ g, float* lds, int lane) {
  const int r  = lane >> 1;
  const int c0 = (lane & 1) * 8;
  const float4* p = (const float4*)(g + r * DD + c0);
  float4 a = p[0];
  float4 b = p[1];
  float* d = lds + r * LDSP + c0;
  d[0] = a.x; d[1] = a.y; d[2] = a.z; d[3] = a.w;
  d[4] = b.x; d[5] = b.y; d[6] = b.z; d[7] = b.w;
}

// A fragment (16x4 slice, M x K) of V_WMMA_F32_16X16X4_F32.
// ISA layout: lanes 0-15 hold M=0-15; VGPR0 holds K = 2*(lane/16), VGPR1 holds K+1.
// Source tile stored row-major [m][k] in LDS.
__device__ __forceinline__ v2f fragA(const float* lds, int lane, int k0) {
  const int m  = lane & 15;
  const int kk = k0 + ((lane >> 4) << 1);
  v2f f;
  f.x = lds[m * LDSP + kk];
  f.y = lds[m * LDSP + kk + 1];
  return f;
}

// B fragment (4x16 slice, K x N) where the LDS tile is stored [n][k] (B = tile^T),
// i.e. for scores = Q * K^T the K-matrix tile is stored row-major [t][d].
__device__ __forceinline__ v2f fragB_T(const float* lds, int lane, int k0) {
  const int n  = lane & 15;
  const int kk = k0 + ((lane >> 4) << 1);
  v2f f;
  f.x = lds[n * LDSP + kk];
  f.y = lds[n * LDSP + kk + 1];
  return f;
}

// B fragment where the LDS tile is stored [k][n] (row-major in the contraction dim).
__device__ __forceinline__ v2f fragB(const float* lds, int lane, int k0) {
  const int n  = lane & 15;
  const int kk = k0 + ((lane >> 4) << 1);
  v2f f;
  f.x = lds[kk * LDSP + n];
  f.y = lds[(kk + 1) * LDSP + n];
  return f;
}

// ---------------------------------------------------------------------------
// Kernel 1: Q/K/V projections. [B*S,16] x [16,16] -> three [B*S,16] matrices.
// Memory-bound and tiny (0.2 GFLOP): plain VALU through LDS.
// ---------------------------------------------------------------------------
__global__ void __launch_bounds__(256)
proj_qkv(const float* __restrict__ x,
         const float* __restrict__ Wq, const float* __restrict__ Wk,
         const float* __restrict__ Wv,
         float* __restrict__ Qg, float* __restrict__ Kg, float* __restrict__ Vg) {
  __shared__ float sx[256];
  __shared__ float swq[256], swk[256], swv[256];
  const int tid = threadIdx.x;
  swq[tid] = Wq[tid];
  swk[tid] = Wk[tid];
  swv[tid] = Wv[tid];
  const int r0 = blockIdx.x * 16; // 16 rows of x per block
  sx[tid] = x[r0 * DD + tid];     // 256 contiguous floats, coalesced
  __syncthreads();
  const int r = tid >> 4, c = tid & 15;
  float q = 0.f, k = 0.f, v = 0.f;
#pragma unroll
  for (int d = 0; d < DD; ++d) {
    const float xv = sx[r * DD + d];
    q = fmaf(xv, swq[d * DD + c], q);
    k = fmaf(xv, swk[d * DD + c], k);
    v = fmaf(xv, swv[d * DD + c], v);
  }
  const int o = (r0 + r) * DD + c;
  Qg[o] = q; Kg[o] = k; Vg[o] = v;
}

// ---------------------------------------------------------------------------
// Kernel 2: per-(s,t)-tile online softmax statistics over the BATCH axis.
// One wave owns a 16x16 (s,t) tile, loops b=0..127, recomputes the score tile
// with V_WMMA_F32_16X16X4_F32 and maintains running max m[s,t] (log2 domain)
// and denominator den = sum_b exp2(x - m) in registers. Stores m and 1/den.
// Single-exp online update: one of exp2(m-nm)/exp2(x-nm) is always 1.
// ---------------------------------------------------------------------------
__global__ void __launch_bounds__(32)
attn_pass1(const float* __restrict__ Qg, const float* __restrict__ Kg,
           float* __restrict__ Mg, float* __restrict__ Rg) {
  __shared__ float qT[16 * LDSP];
  __shared__ float kT[16 * LDSP];
  const int lane = threadIdx.x;
  const int hi = lane >> 4, lo = lane & 15;
  const int s0 = blockIdx.x * 16;
  const int t0 = blockIdx.y * 16;

  float m[8], dn[8];
#pragma unroll
  for (int j = 0; j < 8; ++j) { m[j] = -3.0e38f; dn[j] = 0.0f; }

  for (int b = 0; b < BB; ++b) {
    __syncthreads(); // previous iteration's fragment reads done before restage
    stage_tile(Qg + (b * SS + s0) * DD, qT, lane);
    stage_tile(Kg + (b * SS + t0) * DD, kT, lane);
    if (b + 1 < BB) { // pull next tiles L2 -> WGP$ (global_prefetch_b8)
      __builtin_prefetch(Qg + ((b + 1) * SS + s0) * DD, 0, 3);
      __builtin_prefetch(Kg + ((b + 1) * SS + t0) * DD, 0, 3);
    }
    __syncthreads();
    v8f c = {};
#pragma unroll
    for (int k0 = 0; k0 < DD; k0 += 4) {
      v2f a  = fragA(qT, lane, k0);      // Q[s][d]
      v2f bb = fragB_T(kT, lane, k0);    // (K^T)[d][t]
      c = __builtin_amdgcn_wmma_f32_16x16x4_f32(false, a, false, bb,
                                                (short)0, c, false, false);
    }
    // online max/sum over batch axis, one v_exp per element
#pragma unroll
    for (int j = 0; j < 8; ++j) {
      const float xv = c[j] * SCALE;        // log2-domain logit
      const float d  = xv - m[j];
      const float e  = EXP2F(-fabsf(d));    // exp2(-|d|), single trans op
      const bool  up = d > 0.0f;
      dn[j] = fmaf(dn[j], up ? e : 1.0f, up ? 1.0f : e);
      m[j]  = fmaxf(m[j], xv);
    }
  }
#pragma unroll
  for (int j = 0; j < 8; ++j) {
    const int s = s0 + j + 8 * hi; // C layout: M = vgpr + 8*(lane/16)
    const int t = t0 + lo;         //           N = lane%16
    Mg[s * SS + t] = m[j];
    Rg[s * SS + t] = 1.0f / dn[j]; // reciprocal: pass 2 multiplies instead of divides
  }
}

// ---------------------------------------------------------------------------
// Kernel 3: z[b, s_tile, :] = sum_t exp2(x-m)*rden * V[b, t, :].
// One wave per (s_tile, b): recompute score tile (4 WMMAs), build P in padded
// LDS to reshape C-layout -> A-layout, then P @ V-tile (4 WMMAs).
// ---------------------------------------------------------------------------
__global__ void __launch_bounds__(32)
attn_pass2(const float* __restrict__ Qg, const float* __restrict__ Kg,
           const float* __restrict__ Vg, const float* __restrict__ Mg,
           const float* __restrict__ Rg, float* __restrict__ Og) {
  __shared__ float qT[16 * LDSP];
  __shared__ float kT[16 * LDSP];
  __shared__ float vT[16 * LDSP];
  __shared__ float pT[16 * LDSP];
  const int lane = threadIdx.x;
  const int hi = lane >> 4, lo = lane & 15;
  const int s0 = blockIdx.x * 16;
  const int b  = blockIdx.y;

  stage_tile(Qg + (b * SS + s0) * DD, qT, lane); // reused across all t tiles
  v8f z = {};
  for (int t0 = 0; t0 < SS; t0 += 16) {
    __syncthreads(); // prior-iteration kT/vT/pT reads complete (also covers qT stage)
    stage_tile(Kg + (b * SS + t0) * DD, kT, lane);
    stage_tile(Vg + (b * SS + t0) * DD, vT, lane);
    if (t0 + 16 < SS) {
      __builtin_prefetch(Kg + (b * SS + t0 + 16) * DD, 0, 3);
      __builtin_prefetch(Vg + (b * SS + t0 + 16) * DD, 0, 3);
    }
    __syncthreads();
    v8f c = {};
#pragma unroll
    for (int k0 = 0; k0 < DD; k0 += 4) {
      v2f a  = fragA(qT, lane, k0);
      v2f bb = fragB_T(kT, lane, k0);
      c = __builtin_amdgcn_wmma_f32_16x16x4_f32(false, a, false, bb,
                                                (short)0, c, false, false);
    }
#pragma unroll
    for (int j = 0; j < 8; ++j) {
      const int s = s0 + j + 8 * hi;
      const int t = t0 + lo;
      const float xv = c[j] * SCALE;                              // log2-domain logit
      const float p  = EXP2F(xv - Mg[s * SS + t]) * Rg[s * SS + t]; // L2-resident
      pT[(j + 8 * hi) * LDSP + lo] = p;
    }
    __syncthreads();
#pragma unroll
    for (int k0 = 0; k0 < 16; k0 += 4) {
      v2f a  = fragA(pT, lane, k0); // P[s][t]
      v2f bb = fragB(vT, lane, k0); // V[t][d]
      z = __builtin_amdgcn_wmma_f32_16x16x4_f32(false, a, false, bb,
                                                (short)0, z, false, false);
    }
  }
#pragma unroll
  for (int j = 0; j < 8; ++j) {
    const int s = s0 + j + 8 * hi;
    Og[(b * SS + s) * DD + lo] = z[j];
  }
}

// ---------------------------------------------------------------------------
// Launcher. Workspace layout (floats):
//   Q [B*S*16] | K [B*S*16] | V [B*S*16] | m [S*S] | 1/den [S*S]  = 32 MB total
// ---------------------------------------------------------------------------
extern "C" void kernel_launch(void* const* d_in, const int* in_sizes, int n_in,
                              void* d_out, int out_size, void* d_ws, size_t ws_size,
                              hipStream_t stream) {
  (void)in_sizes; (void)n_in; (void)out_size; (void)ws_size;
  const float* x  = (const float*)d_in[0];
  const float* Wq = (const float*)d_in[1];
  const float* Wk = (const float*)d_in[2];
  const float* Wv = (const float*)d_in[3];

  float* ws = (float*)d_ws;
  const size_t qkv = (size_t)BB * SS * DD; // 2,097,152 floats
  float* Q = ws;
  float* K = ws + qkv;
  float* V = ws + 2 * qkv;
  float* M = ws + 3 * qkv;                  // [S,S] running max (log2 domain)
  float* R = M + (size_t)SS * SS;           // [S,S] reciprocal denominator

  proj_qkv<<<dim3((BB * SS) / 16), dim3(256), 0, stream>>>(x, Wq, Wk, Wv, Q, K, V);
  attn_pass1<<<dim3(SS / 16, SS / 16), dim3(32), 0, stream>>>(Q, K, M, R);
  attn_pass2<<<dim3(SS / 16, BB), dim3(32), 0, stream>>>(Q, K, V, M, R, (float*)d_out);
}